// DMPNet_49838800503039
// MI455X (gfx1250) — compile-verified
//
#include <hip/hip_runtime.h>
#include <math.h>

// ---------------- problem constants (from reference) ----------------
#define B_TOT   65536
#define KDIM    256
#define DOFN    7
#define NB      5        // RBF basis count
#define NOUT    42       // DOF*(N+1)
#define NPAD    48       // padded GEMM N (3 x 16 WMMA col tiles)
#define NSAMP   11       // sampled time points (0,10,...,100)
#define NSTEP   100
#define DTC     0.01f    // TAU/(T*L)
#define AZ_C    15.0f
#define BZ_C    3.75f    // A_Z/4
#define Z0_C    0.05f    // dy0*TAU

typedef float v2f __attribute__((ext_vector_type(2)));
typedef float v8f __attribute__((ext_vector_type(8)));

// -------------------------------------------------------------------
// Kernel 1: DMP linear-coefficient table.
// The DMP recurrence is linear in basis e = (y0, z0, g, q0..q4) with
// q_n = w_n*(g - y0).  Track coefficient vectors of y and z over this
// basis for 100 steps; record y-coefficients at steps 0,10,...,100.
// coef layout: coef[s*8 + i], s = 0..10.
// -------------------------------------------------------------------
__global__ void dmp_coef_kernel(float* __restrict__ coef) {
  if (threadIdx.x != 0 || blockIdx.x != 0) return;
  float c[NB], s2[NB];
  for (int n = 0; n < NB; ++n) {
    c[n]  = expf(-0.25f * (float)n);            // exp(-A_X*linspace(0,1,5))
    s2[n] = 11.180339887498949f / c[n];         // N^1.5 / c / A_X
  }
  float yc[8] = {1.f, 0.f, 0.f, 0.f, 0.f, 0.f, 0.f, 0.f};
  float zc[8] = {0.f, 1.f, 0.f, 0.f, 0.f, 0.f, 0.f, 0.f};
  for (int i = 0; i < 8; ++i) coef[i] = yc[i];  // sample s=0 (y = y0)

  float x = 1.0f;
  for (int t = 1; t <= NSTEP; ++t) {
    x = x - x * DTC;                            // x += (-A_X*x*TAU)*DT
    float psi[NB], S = 0.0f;
    for (int n = 0; n < NB; ++n) {
      float d = x - c[n];
      psi[n] = expf(-0.5f * d * d / s2[n]);
      S += psi[n];
    }
    float fc[8] = {0.f, 0.f, 0.f, 0.f, 0.f, 0.f, 0.f, 0.f};
    for (int n = 0; n < NB; ++n) fc[3 + n] = psi[n] * x / S;   // phi_t[n]

    float dyc[8], dzc[8];
    for (int i = 0; i < 8; ++i) {
      float eg = (i == 2) ? 1.0f : 0.0f;        // goal basis component
      dyc[i] = zc[i];                            // * TAU (=1)
      dzc[i] = AZ_C * (BZ_C * (eg - yc[i]) - zc[i]) + fc[i];
    }
    for (int i = 0; i < 8; ++i) { yc[i] += dyc[i] * DTC; zc[i] += dzc[i] * DTC; }
    if (t % 10 == 0) {
      float* row = coef + (t / 10) * 8;
      for (int i = 0; i < 8; ++i) row[i] = yc[i];
    }
  }
}

// -------------------------------------------------------------------
// Kernel 2: transpose W[256,42] -> Wt[48][256] (zero-padded cols) so
// WMMA B-fragments are contiguous b64 loads per lane.
// -------------------------------------------------------------------
__global__ void wt_kernel(const float* __restrict__ W, float* __restrict__ wt) {
  int i = blockIdx.x * blockDim.x + threadIdx.x;
  if (i >= NPAD * KDIM) return;
  int n = i / KDIM;
  int k = i - n * KDIM;
  wt[i] = (n < NOUT) ? W[k * NOUT + n] : 0.0f;
}

// -------------------------------------------------------------------
// Epilogue for one 16-row half: stage C tiles to LDS, evaluate the
// bilinear DMP closed form per (row,dof) channel, stage [16][77] and
// store contiguously.
// -------------------------------------------------------------------
__device__ __forceinline__ void epilogue16(
    float* __restrict__ s42, float* __restrict__ sy,
    v8f a0, v8f a1, v8f a2,
    int lane, int half, int mrow, int rb16,
    const float* __restrict__ state, const float* __restrict__ bias,
    const float* __restrict__ coef, float* __restrict__ out) {
  // ---- stage GEMM tile into LDS (C layout: lane = N, vgpr r -> M) ----
#pragma unroll
  for (int r = 0; r < 8; ++r) {
    int row = r + 8 * half;
    s42[row * NPAD + mrow +  0] = a0[r];
    s42[row * NPAD + mrow + 16] = a1[r];
    s42[row * NPAD + mrow + 32] = a2[r];
  }
  // same-wave DS ops complete in order: staged data visible below.

  // ---- DMP evaluation: 112 channels (16 rows x 7 dof) ----
  for (int ch = lane; ch < 16 * DOFN; ch += 32) {
    int row = ch / DOFN;
    int d   = ch - row * DOFN;
    float g  = s42[row * NPAD + d] + bias[d];
    float y0 = state[(size_t)(rb16 + row) * DOFN + d];
    int wb = 7 + 5 * d;
    float w0 = s42[row * NPAD + wb + 0] + bias[wb + 0];
    float w1 = s42[row * NPAD + wb + 1] + bias[wb + 1];
    float w2 = s42[row * NPAD + wb + 2] + bias[wb + 2];
    float w3 = s42[row * NPAD + wb + 3] + bias[wb + 3];
    float w4 = s42[row * NPAD + wb + 4] + bias[wb + 4];
    float gm = g - y0;
#pragma unroll
    for (int s = 0; s < NSAMP; ++s) {
      const float* cs = coef + s * 8;
      float v = cs[0] * y0 + cs[1] * Z0_C + cs[2] * g
              + gm * (cs[3] * w0 + cs[4] * w1 + cs[5] * w2
                    + cs[6] * w3 + cs[7] * w4);
      sy[row * 77 + s * DOFN + d] = v;
    }
  }

  // ---- fully coalesced contiguous store of 16 output rows ----
  float* orow = out + (size_t)rb16 * 77;
  for (int i = lane; i < 16 * 77; i += 32) orow[i] = sy[i];
}

// -------------------------------------------------------------------
// Kernel 3: fused GEMM (f32 WMMA 16x16x4) + DMP evaluation + output.
// 8 waves per block; each wave owns a 32-row tile (two A fragments
// share three B fragments: 5 loads per 6 WMMAs). 256 blocks total.
// -------------------------------------------------------------------
__global__ __launch_bounds__(256) void fused_kernel(
    const float* __restrict__ x, const float* __restrict__ state,
    const float* __restrict__ wt, const float* __restrict__ bias,
    const float* __restrict__ coef, float* __restrict__ out) {
  __shared__ float smem[8 * 2048];   // 64 KB: per-wave 2048 floats
  const int lane = threadIdx.x & 31;
  const int wave = threadIdx.x >> 5;
  const int row_base = (blockIdx.x * 8 + wave) * 32;

  float* s42 = smem + wave * 2048;   // [16][48] staged GEMM half-tile
  float* sy  = s42 + 16 * NPAD;      // [16][77] staged final outputs

  const int half = lane >> 4;        // 0: K={k,k+1}, 1: K={k+2,k+3}
  const int mrow = lane & 15;        // A: M row   B: N col

  // ---- GEMM: 32x256 @ 256x48 via 6 accumulators ----
  v8f acc00 = {}; v8f acc01 = {}; v8f acc02 = {};   // rows  0..15
  v8f acc10 = {}; v8f acc11 = {}; v8f acc12 = {};   // rows 16..31
  const float* ap0 = x  + (size_t)(row_base + mrow) * KDIM + 2 * half;
  const float* ap1 = ap0 + 16 * KDIM;
  const float* bp0 = wt + (size_t)(mrow +  0) * KDIM + 2 * half;
  const float* bp1 = wt + (size_t)(mrow + 16) * KDIM + 2 * half;
  const float* bp2 = wt + (size_t)(mrow + 32) * KDIM + 2 * half;
#pragma unroll 4
  for (int k = 0; k < KDIM; k += 4) {
    v2f a0 = *(const v2f*)(ap0 + k);
    v2f a1 = *(const v2f*)(ap1 + k);
    v2f b0 = *(const v2f*)(bp0 + k);
    v2f b1 = *(const v2f*)(bp1 + k);
    v2f b2 = *(const v2f*)(bp2 + k);
    acc00 = __builtin_amdgcn_wmma_f32_16x16x4_f32(false, a0, false, b0, (short)0, acc00, false, false);
    acc01 = __builtin_amdgcn_wmma_f32_16x16x4_f32(false, a0, false, b1, (short)0, acc01, false, false);
    acc02 = __builtin_amdgcn_wmma_f32_16x16x4_f32(false, a0, false, b2, (short)0, acc02, false, false);
    acc10 = __builtin_amdgcn_wmma_f32_16x16x4_f32(false, a1, false, b0, (short)0, acc10, false, false);
    acc11 = __builtin_amdgcn_wmma_f32_16x16x4_f32(false, a1, false, b1, (short)0, acc11, false, false);
    acc12 = __builtin_amdgcn_wmma_f32_16x16x4_f32(false, a1, false, b2, (short)0, acc12, false, false);
  }

  // ---- epilogue: two 16-row halves reuse the same 8 KB LDS region ----
  epilogue16(s42, sy, acc00, acc01, acc02, lane, half, mrow, row_base,
             state, bias, coef, out);
  epilogue16(s42, sy, acc10, acc11, acc12, lane, half, mrow, row_base + 16,
             state, bias, coef, out);
}

// -------------------------------------------------------------------
extern "C" void kernel_launch(void* const* d_in, const int* in_sizes, int n_in,
                              void* d_out, int out_size, void* d_ws, size_t ws_size,
                              hipStream_t stream) {
  const float* x     = (const float*)d_in[0];   // [65536,256]
  const float* state = (const float*)d_in[1];   // [65536,7]
  const float* W     = (const float*)d_in[2];   // [256,42]
  const float* bias  = (const float*)d_in[3];   // [42]
  float* out = (float*)d_out;

  float* ws   = (float*)d_ws;
  float* coef = ws;            // 88 floats (11x8 coefficient table)
  float* wt   = ws + 128;      // 48*256 floats, transposed/padded W

  dmp_coef_kernel<<<1, 32, 0, stream>>>(coef);
  wt_kernel<<<(NPAD * KDIM + 255) / 256, 256, 0, stream>>>(W, wt);
  fused_kernel<<<B_TOT / 256, 256, 0, stream>>>(x, state, wt, bias, coef, out);
}